// GATLayer_65687229825553
// MI455X (gfx1250) — compile-verified
//
#include <hip/hip_runtime.h>
#include <hip/hip_bf16.h>
#include <math.h>

typedef float v2f __attribute__((ext_vector_type(2)));
typedef float v8f __attribute__((ext_vector_type(8)));

#define HEADS 4
#define DH    32
#define DM    128   // D_IN == D_OUT == 128

// ---- order-preserving float <-> uint map for atomic segment-max ----
__device__ __forceinline__ unsigned mono_enc(float f) {
  unsigned u = __float_as_uint(f);
  return (u & 0x80000000u) ? ~u : (u | 0x80000000u);
}
__device__ __forceinline__ float mono_dec(unsigned u) {
  return (u & 0x80000000u) ? __uint_as_float(u & 0x7FFFFFFFu)
                           : __uint_as_float(~u);
}

// ---------------------------------------------------------------------------
// C[M x 128] = A[M x 128] * B[128 x 128], fp32 WMMA 16x16x4, one wave per tile
// A layout (16x4 f32): lanes 0-15 -> M=lane, V0=K0/V1=K1; lanes 16-31 -> K2/K3
// B layout (4x16 f32): lanes 0-15 -> N=lane, V0=K0/V1=K1; lanes 16-31 -> K2/K3
// C/D layout: VGPR r: lanes 0-15 -> M=r, lanes 16-31 -> M=r+8 (N=lane&15)
// ---------------------------------------------------------------------------
__global__ __launch_bounds__(256) void gemm_wx_wmma(
    const float* __restrict__ A, const float* __restrict__ B,
    float* __restrict__ C, int M)
{
  const int wave = (int)((blockIdx.x * blockDim.x + threadIdx.x) >> 5);
  const int mtiles = M >> 4;
  const int mt = wave >> 3;          // 128/16 = 8 column tiles
  const int nt = wave & 7;
  if (mt >= mtiles) return;          // wave-uniform exit: EXEC stays all-ones
  const int lane = threadIdx.x & 31;
  const int lo = lane & 15;
  const int hi = lane >> 4;          // selects K pair {0,1} vs {2,3}
  const int arow = (mt << 4) + lo;
  const int bcol = (nt << 4) + lo;

  v8f c = {};
  for (int k0 = 0; k0 < DM; k0 += 4) {
    const int ka = k0 + (hi << 1);
    v2f a, b;
    a.x = A[arow * DM + ka];
    a.y = A[arow * DM + ka + 1];
    b.x = B[ka * DM + bcol];
    b.y = B[(ka + 1) * DM + bcol];
    c = __builtin_amdgcn_wmma_f32_16x16x4_f32(false, a, false, b,
                                              (short)0, c, false, false);
  }
  const int crow0 = (mt << 4) + (hi << 3);
#pragma unroll
  for (int r = 0; r < 8; ++r)
    C[(crow0 + r) * DM + bcol] = c[r];
}

// Same GEMM with fused bias + ELU epilogue (final projection)
__global__ __launch_bounds__(256) void gemm_out_wmma(
    const float* __restrict__ A, const float* __restrict__ B,
    const float* __restrict__ bias, float* __restrict__ O, int M)
{
  const int wave = (int)((blockIdx.x * blockDim.x + threadIdx.x) >> 5);
  const int mtiles = M >> 4;
  const int mt = wave >> 3;
  const int nt = wave & 7;
  if (mt >= mtiles) return;
  const int lane = threadIdx.x & 31;
  const int lo = lane & 15;
  const int hi = lane >> 4;
  const int arow = (mt << 4) + lo;
  const int bcol = (nt << 4) + lo;

  v8f c = {};
  for (int k0 = 0; k0 < DM; k0 += 4) {
    const int ka = k0 + (hi << 1);
    v2f a, b;
    a.x = A[arow * DM + ka];
    a.y = A[arow * DM + ka + 1];
    b.x = B[ka * DM + bcol];
    b.y = B[(ka + 1) * DM + bcol];
    c = __builtin_amdgcn_wmma_f32_16x16x4_f32(false, a, false, b,
                                              (short)0, c, false, false);
  }
  const float bb = bias[bcol];
  const int crow0 = (mt << 4) + (hi << 3);
#pragma unroll
  for (int r = 0; r < 8; ++r) {
    float v = c[r] + bb;
    v = (v > 0.f) ? v : (expf(v) - 1.f);   // ELU
    O[(crow0 + r) * DM + bcol] = v;
  }
}

// per-node attention scores: s[(n*H+h)*2+0] = <Wx[n,h,:], a_i[h]>, +1 -> a_j
__global__ __launch_bounds__(256) void node_scores(
    const float* __restrict__ Wx, const float* __restrict__ attn,
    float* __restrict__ s, int N)
{
  const int t = blockIdx.x * blockDim.x + threadIdx.x;
  if (t >= N * HEADS) return;
  const int node = t >> 2, h = t & 3;
  const float* v  = Wx + (size_t)node * DM + h * DH;
  const float* ai = attn + h * (2 * DH);
  float si = 0.f, sj = 0.f;
#pragma unroll
  for (int d = 0; d < DH; ++d) {
    si += v[d] * ai[d];
    sj += v[d] * ai[DH + d];
  }
  s[t * 2 + 0] = si;
  s[t * 2 + 1] = sj;
}

// edge logits + leaky-relu; store; atomic segment-max over dst
__global__ __launch_bounds__(256) void edge_logits(
    const long long* __restrict__ ei, const float* __restrict__ s,
    float* __restrict__ e_edge, unsigned* __restrict__ emax, int E)
{
  const int t = blockIdx.x * blockDim.x + threadIdx.x;
  if (t >= E * HEADS) return;
  const int e = t >> 2, h = t & 3;
  const int src = (int)ei[e];
  const int dst = (int)ei[E + e];
  float v = s[(src * HEADS + h) * 2] + s[(dst * HEADS + h) * 2 + 1];
  v = (v > 0.f) ? v : 0.2f * v;            // leaky_relu(0.2)
  e_edge[t] = v;
  atomicMax(&emax[dst * HEADS + h], mono_enc(v));
}

// exp(e - max) in place; atomic segment-sum over dst
__global__ __launch_bounds__(256) void edge_expsum(
    const long long* __restrict__ ei, float* __restrict__ e_edge,
    const unsigned* __restrict__ emax, float* __restrict__ esum, int E)
{
  const int t = blockIdx.x * blockDim.x + threadIdx.x;
  if (t >= E * HEADS) return;
  const int e = t >> 2, h = t & 3;
  const int dst = (int)ei[E + e];
  const float m = mono_dec(emax[dst * HEADS + h]);
  const float ex = expf(e_edge[t] - m);
  e_edge[t] = ex;
  atomicAdd(&esum[dst * HEADS + h], ex);
}

// one wave per edge: lane owns 4 features (float4 gather, 4 atomic adds)
__global__ __launch_bounds__(256) void edge_aggregate(
    const long long* __restrict__ ei, const float* __restrict__ Wx,
    const float* __restrict__ e_edge, const float* __restrict__ esum,
    float* __restrict__ agg, int E)
{
  const int eid = (int)((blockIdx.x * blockDim.x + threadIdx.x) >> 5);
  if (eid >= E) return;                    // wave-uniform
  const int lane = threadIdx.x & 31;
  const int h = lane >> 3;                 // 8 lanes (32 feats) per head
  const int src = (int)ei[eid];
  const int dst = (int)ei[E + eid];
  const float denom = esum[dst * HEADS + h] + 1e-16f;
  const float alpha = e_edge[eid * HEADS + h] / denom;
  const float4 hv = *(const float4*)(Wx + (size_t)src * DM + lane * 4);
  float* o = agg + (size_t)dst * DM + lane * 4;
  atomicAdd(o + 0, alpha * hv.x);
  atomicAdd(o + 1, alpha * hv.y);
  atomicAdd(o + 2, alpha * hv.z);
  atomicAdd(o + 3, alpha * hv.w);
}

extern "C" void kernel_launch(void* const* d_in, const int* in_sizes, int n_in,
                              void* d_out, int out_size, void* d_ws, size_t ws_size,
                              hipStream_t stream)
{
  const float*     x     = (const float*)d_in[0];
  const long long* ei    = (const long long*)d_in[1];   // int64 per reference
  const float*     W     = (const float*)d_in[2];
  const float*     attn  = (const float*)d_in[3];
  const float*     out_w = (const float*)d_in[4];
  const float*     out_b = (const float*)d_in[5];
  float*           out   = (float*)d_out;

  const int N = in_sizes[0] / DM;      // 50000 (divisible by 16)
  const int E = in_sizes[1] / 2;       // 800000

  // workspace layout (floats): Wx | s | e_edge | emax(u32) | esum | agg
  float*    Wx     = (float*)d_ws;
  float*    s      = Wx + (size_t)N * DM;
  float*    e_edge = s + (size_t)N * HEADS * 2;
  unsigned* emax   = (unsigned*)(e_edge + (size_t)E * HEADS);
  float*    esum   = (float*)(emax + (size_t)N * HEADS);
  float*    agg    = esum + (size_t)N * HEADS;

  // zero the accumulators (emax | esum | agg are contiguous) every call
  hipMemsetAsync(emax, 0,
                 ((size_t)N * HEADS * 2 + (size_t)N * DM) * sizeof(float),
                 stream);

  // 1) Wx = x @ W  (WMMA fp32)
  {
    const int waves  = (N / 16) * (DM / 16);
    const int blocks = (waves + 7) / 8;               // 8 waves / 256-thr block
    gemm_wx_wmma<<<blocks, 256, 0, stream>>>(x, W, Wx, N);
  }
  // 2) per-node scores
  {
    const int threads = N * HEADS;
    node_scores<<<(threads + 255) / 256, 256, 0, stream>>>(Wx, attn, s, N);
  }
  // 3) edge logits + segment max
  {
    const int threads = E * HEADS;
    edge_logits<<<(threads + 255) / 256, 256, 0, stream>>>(ei, s, e_edge, emax, E);
  }
  // 4) exp + segment sum
  {
    const int threads = E * HEADS;
    edge_expsum<<<(threads + 255) / 256, 256, 0, stream>>>(ei, e_edge, emax, esum, E);
  }
  // 5) weighted scatter-aggregate (one wave per edge)
  {
    const long long threads = (long long)E * 32;
    edge_aggregate<<<(int)((threads + 255) / 256), 256, 0, stream>>>(
        ei, Wx, e_edge, esum, agg, E);
  }
  // 6) out = elu(agg @ out_w + b)  (WMMA fp32, fused epilogue)
  {
    const int waves  = (N / 16) * (DM / 16);
    const int blocks = (waves + 7) / 8;
    gemm_out_wmma<<<blocks, 256, 0, stream>>>(agg, out_w, out_b, out, N);
  }
}